// GNNDecoder_56659208569397
// MI455X (gfx1250) — compile-verified
//
#include <hip/hip_runtime.h>
#include <hip/hip_bf16.h>

typedef __attribute__((ext_vector_type(2))) float v2f;
typedef __attribute__((ext_vector_type(8))) float v8f;

#define HD 128            // hidden dim
#define MID 256           // 2*H
#define WAVES 8           // waves per block (256 threads, wave32)

// D = A(16x4 f32) * B(4x16 f32) + C, full fp32 WMMA
__device__ __forceinline__ v8f wmma_f32_k4(v2f a, v2f b, v8f c) {
    return __builtin_amdgcn_wmma_f32_16x16x4_f32(
        /*neg_a=*/false, a, /*neg_b=*/false, b,
        /*c_mod=*/(short)0, c, /*reuse_a=*/false, /*reuse_b=*/false);
}

// ---------------------------------------------------------------------------
// Kernel 1: Y = prelu(X) @ Wenc     (N x 128) @ (128 x 128)
// One block = 16-row strip; 8 waves each own one 16x16 column tile.
// ---------------------------------------------------------------------------
__global__ __launch_bounds__(256) void k_prelu_gemm(
    const float* __restrict__ X, const float* __restrict__ Wenc,
    const float* __restrict__ pa, float* __restrict__ Y, int N)
{
    __shared__ float Xs[16 * HD];          // 8 KB
    const int row0 = blockIdx.x * 16;
    const int tid  = threadIdx.x;
    const float a  = pa[0];

    // cooperative load + PReLU; clamp row for tail safety
    for (int i = tid; i < 16 * HD / 4; i += 256) {
        int r  = i >> 5;                   // 32 float4 per row
        int c4 = i & 31;
        int gr = row0 + r; if (gr >= N) gr = N - 1;
        float4 v = ((const float4*)(X + (size_t)gr * HD))[c4];
        v.x = v.x > 0.f ? v.x : a * v.x;
        v.y = v.y > 0.f ? v.y : a * v.y;
        v.z = v.z > 0.f ? v.z : a * v.z;
        v.w = v.w > 0.f ? v.w : a * v.w;
        ((float4*)Xs)[i] = v;
    }
    __syncthreads();

    const int wave  = tid >> 5;
    const int lane  = tid & 31;
    const int lrow  = lane & 15;
    const int khalf = (lane >> 4) * 2;     // 0 or 2
    const int m0    = (lane >> 4) * 8;
    const int col   = wave * 16;

    v8f acc = {};
    #pragma unroll
    for (int k = 0; k < HD; k += 4) {
        v2f af, bf;
        af.x = Xs[lrow * HD + k + khalf];
        af.y = Xs[lrow * HD + k + khalf + 1];
        bf.x = Wenc[(k + khalf)     * HD + col + lrow];
        bf.y = Wenc[(k + khalf + 1) * HD + col + lrow];
        acc = wmma_f32_k4(af, bf, acc);
    }
    float* op = Y + (size_t)row0 * HD + col;
    #pragma unroll
    for (int v = 0; v < 8; ++v)
        if (row0 + m0 + v < N) op[(size_t)(m0 + v) * HD + lrow] = acc[v];
}

// ---------------------------------------------------------------------------
// Kernel 2a: agg[i] = Y[i] + emb1[5] + emb2[0]   (self-loop folded in, no atomics)
// ---------------------------------------------------------------------------
__global__ __launch_bounds__(256) void k_agg_init(
    const float* __restrict__ Y, const float* __restrict__ e1,
    const float* __restrict__ e2, float* __restrict__ agg, int total)
{
    int i = blockIdx.x * 256 + threadIdx.x;
    if (i < total) {
        int c = i & (HD - 1);
        agg[i] = Y[i] + e1[5 * HD + c] + e2[c];
    }
}

// ---------------------------------------------------------------------------
// Kernel 2b: agg[dst] += Y[src] + emb1[a1] + emb2[a2]  (one wave per edge)
// ---------------------------------------------------------------------------
__global__ __launch_bounds__(256) void k_scatter(
    const int* __restrict__ src, const int* __restrict__ dst,
    const int* __restrict__ eattr,
    const float* __restrict__ Y, const float* __restrict__ e1,
    const float* __restrict__ e2, float* __restrict__ agg, int E)
{
    int g    = blockIdx.x * 256 + threadIdx.x;
    int edge = g >> 5;
    int lane = g & 31;
    if (edge >= E) return;

    int s  = src[edge];
    int d  = dst[edge];
    int a1 = eattr[edge * 2 + 0];
    int a2 = eattr[edge * 2 + 1];
    int c  = lane * 4;

    float4 v  = *(const float4*)(Y  + (size_t)s  * HD + c);
    float4 w1 = *(const float4*)(e1 + (size_t)a1 * HD + c);
    float4 w2 = *(const float4*)(e2 + (size_t)a2 * HD + c);
    float* p  = agg + (size_t)d * HD + c;
    unsafeAtomicAdd(p + 0, v.x + w1.x + w2.x);   // global_atomic_add_f32
    unsafeAtomicAdd(p + 1, v.y + w1.y + w2.y);
    unsafeAtomicAdd(p + 2, v.z + w1.z + w2.z);
    unsafeAtomicAdd(p + 3, v.w + w1.w + w2.w);
}

// ---------------------------------------------------------------------------
// Kernel 3: OUT = silu(AGG @ W1 + b1) @ W2 + b2, fused through LDS
// ---------------------------------------------------------------------------
__global__ __launch_bounds__(256) void k_mlp(
    const float* __restrict__ AGG,
    const float* __restrict__ W1, const float* __restrict__ b1,
    const float* __restrict__ W2, const float* __restrict__ b2,
    float* __restrict__ OUT, int N)
{
    __shared__ float As[16 * HD];          // 8 KB
    __shared__ float Hs[16 * MID];         // 16 KB
    const int row0 = blockIdx.x * 16;
    const int tid  = threadIdx.x;

    for (int i = tid; i < 16 * HD / 4; i += 256) {
        int r  = i >> 5;
        int c4 = i & 31;
        int gr = row0 + r; if (gr >= N) gr = N - 1;
        ((float4*)As)[i] = ((const float4*)(AGG + (size_t)gr * HD))[c4];
    }
    __syncthreads();

    const int wave  = tid >> 5;
    const int lane  = tid & 31;
    const int lrow  = lane & 15;
    const int khalf = (lane >> 4) * 2;
    const int m0    = (lane >> 4) * 8;

    // Phase 1: H = silu(A @ W1 + b1); 16 column tiles, 2 per wave
    #pragma unroll
    for (int t = 0; t < 2; ++t) {
        const int col = (wave + t * WAVES) * 16;
        v8f acc = {};
        #pragma unroll
        for (int k = 0; k < HD; k += 4) {
            v2f af, bf;
            af.x = As[lrow * HD + k + khalf];
            af.y = As[lrow * HD + k + khalf + 1];
            bf.x = W1[(k + khalf)     * MID + col + lrow];
            bf.y = W1[(k + khalf + 1) * MID + col + lrow];
            acc = wmma_f32_k4(af, bf, acc);
        }
        const float bb = b1[col + lrow];
        #pragma unroll
        for (int v = 0; v < 8; ++v) {
            float z = acc[v] + bb;
            Hs[(m0 + v) * MID + col + lrow] = z / (1.f + __expf(-z));  // SiLU
        }
    }
    __syncthreads();

    // Phase 2: OUT = H @ W2 + b2; 8 column tiles, 1 per wave, K=256
    const int col = wave * 16;
    v8f acc = {};
    #pragma unroll
    for (int k = 0; k < MID; k += 4) {
        v2f af, bf;
        af.x = Hs[lrow * MID + k + khalf];
        af.y = Hs[lrow * MID + k + khalf + 1];
        bf.x = W2[(k + khalf)     * HD + col + lrow];
        bf.y = W2[(k + khalf + 1) * HD + col + lrow];
        acc = wmma_f32_k4(af, bf, acc);
    }
    const float bb = b2[col + lrow];
    float* op = OUT + (size_t)row0 * HD + col;
    #pragma unroll
    for (int v = 0; v < 8; ++v)
        if (row0 + m0 + v < N) op[(size_t)(m0 + v) * HD + lrow] = acc[v] + bb;
}

// ---------------------------------------------------------------------------
extern "C" void kernel_launch(void* const* d_in, const int* in_sizes, int n_in,
                              void* d_out, int out_size, void* d_ws, size_t ws_size,
                              hipStream_t stream) {
    const float* x          = (const float*)d_in[0];
    const int*   edge_index = (const int*)  d_in[1];
    const int*   edge_attr  = (const int*)  d_in[2];
    const float* prelu_a    = (const float*)d_in[3];
    const float* Wenc       = (const float*)d_in[4];
    const float* e1         = (const float*)d_in[5];
    const float* e2         = (const float*)d_in[6];
    const float* W1         = (const float*)d_in[7];
    const float* b1         = (const float*)d_in[8];
    const float* W2         = (const float*)d_in[9];
    const float* b2         = (const float*)d_in[10];
    float* out = (float*)d_out;

    const int N = in_sizes[0] / HD;
    const int E = in_sizes[1] / 2;

    float* Y   = (float*)d_ws;                  // [N,128]  25.6 MB
    float* agg = Y + (size_t)N * HD;            // [N,128]  25.6 MB

    const int nstrips = (N + 15) / 16;

    k_prelu_gemm<<<nstrips, 256, 0, stream>>>(x, Wenc, prelu_a, Y, N);
    k_agg_init  <<<((size_t)N * HD + 255) / 256, 256, 0, stream>>>(Y, e1, e2, agg, N * HD);
    k_scatter   <<<((size_t)E * 32 + 255) / 256, 256, 0, stream>>>(
                    edge_index, edge_index + E, edge_attr, Y, e1, e2, agg, E);
    k_mlp       <<<nstrips, 256, 0, stream>>>(agg, W1, b1, W2, b2, out, N);
}